// BinaryPositionIO_62380105007278
// MI455X (gfx1250) — compile-verified
//
#include <hip/hip_runtime.h>
#include <hip/hip_bf16.h>
#include <math.h>

typedef __attribute__((ext_vector_type(2))) float v2f;
typedef __attribute__((ext_vector_type(4))) float f4;
typedef __attribute__((ext_vector_type(8))) float v8f;

#define BB 32
#define SS 4096
#define DD 512
#define NCHUNK 16
#define SCHUNK (SS / NCHUNK)   // 256

// ---------------------------------------------------------------------------
// Kernel 1: masked softmax over hamming-distance scores -> weights (B x S)
// score = ((12-h)*log1p(eps) + h*log(eps)) / 0.1,  h = popcount(rel ^ q) on 12 bits
// ---------------------------------------------------------------------------
__global__ __launch_bounds__(256) void k_weights(const int* __restrict__ positions,
                                                 const int* __restrict__ anchor,
                                                 const float* __restrict__ read_offset,
                                                 const int* __restrict__ input_length,
                                                 float* __restrict__ w_out)
{
    __shared__ float red[256];
    const int b   = blockIdx.x;
    const int t   = threadIdx.x;
    const int anc = anchor[b];
    const int len = input_length[b];
    const int q   = (int)fminf(fmaxf(read_offset[b], 0.0f), 4095.0f);
    const float lg1 = log1pf(1e-8f);
    const float lg0 = logf(1e-8f);

    // pass 1: masked max
    float mx = -INFINITY;
    for (int s = t; s < SS; s += 256) {
        int pos = positions[b * SS + s];
        if (pos > anc && pos <= anc + len) {
            float relf = fminf(fmaxf((float)pos - (float)anc - 1.0f, 0.0f), 4095.0f);
            int h = __popc(((int)relf ^ q) & 0xFFF);
            float sc = ((float)(12 - h) * lg1 + (float)h * lg0) * 10.0f;
            mx = fmaxf(mx, sc);
        }
    }
    red[t] = mx; __syncthreads();
    for (int o = 128; o > 0; o >>= 1) { if (t < o) red[t] = fmaxf(red[t], red[t + o]); __syncthreads(); }
    mx = red[0]; __syncthreads();

    // pass 2: sum of exp
    float sm = 0.0f;
    for (int s = t; s < SS; s += 256) {
        int pos = positions[b * SS + s];
        if (pos > anc && pos <= anc + len) {
            float relf = fminf(fmaxf((float)pos - (float)anc - 1.0f, 0.0f), 4095.0f);
            int h = __popc(((int)relf ^ q) & 0xFFF);
            float sc = ((float)(12 - h) * lg1 + (float)h * lg0) * 10.0f;
            sm += expf(sc - mx);
        }
    }
    red[t] = sm; __syncthreads();
    for (int o = 128; o > 0; o >>= 1) { if (t < o) red[t] += red[t + o]; __syncthreads(); }
    sm = red[0]; __syncthreads();
    const float inv = 1.0f / sm;

    // pass 3: write normalized weights
    for (int s = t; s < SS; s += 256) {
        int pos = positions[b * SS + s];
        float w = 0.0f;
        if (pos > anc && pos <= anc + len) {
            float relf = fminf(fmaxf((float)pos - (float)anc - 1.0f, 0.0f), 4095.0f);
            int h = __popc(((int)relf ^ q) & 0xFFF);
            float sc = ((float)(12 - h) * lg1 + (float)h * lg0) * 10.0f;
            w = expf(sc - mx) * inv;
        }
        w_out[b * SS + s] = w;
    }
}

// ---------------------------------------------------------------------------
// Kernel 2: streaming reduction  partial[b,c,d] = sum_{s in chunk c} w[b,s]*x[b,s,d]
// 256 MB of x read exactly once, fully coalesced 128-bit loads. HBM-bound.
// ---------------------------------------------------------------------------
__global__ __launch_bounds__(128) void k_wsum(const float* __restrict__ x,
                                              const float* __restrict__ w,
                                              float* __restrict__ partial)
{
    __shared__ float wl[SCHUNK];
    const int b  = blockIdx.y;
    const int c  = blockIdx.x;
    const int t  = threadIdx.x;         // 128 threads, 4 floats each = full D row
    const int s0 = c * SCHUNK;
    for (int i = t; i < SCHUNK; i += 128) wl[i] = w[b * SS + s0 + i];
    __syncthreads();

    f4 acc = {0.0f, 0.0f, 0.0f, 0.0f};
    const float* xp = x + ((size_t)b * SS + s0) * DD + 4 * t;
    for (int i = 0; i < SCHUNK; ++i) {
        f4 xv = *(const f4*)(xp + (size_t)i * DD);
        float wi = wl[i];
        acc.x += wi * xv.x;
        acc.y += wi * xv.y;
        acc.z += wi * xv.z;
        acc.w += wi * xv.w;
    }
    *(f4*)(partial + ((size_t)b * NCHUNK + c) * DD + 4 * t) = acc;
}

// ---------------------------------------------------------------------------
// Kernel 2b: xw[b,d] = sum_c partial[b,c,d]   (deterministic, no float atomics)
// ---------------------------------------------------------------------------
__global__ __launch_bounds__(256) void k_reduce(const float* __restrict__ partial,
                                                float* __restrict__ xw)
{
    int idx = blockIdx.x * 256 + threadIdx.x;   // B*D = 16384
    if (idx >= BB * DD) return;
    int b = idx / DD, d = idx % DD;
    float s = 0.0f;
    #pragma unroll
    for (int c = 0; c < NCHUNK; ++c) s += partial[((size_t)b * NCHUNK + c) * DD + d];
    xw[idx] = s;
}

// ---------------------------------------------------------------------------
// Kernel 3: attended(32x512) = xw(32x512) @ W_value^T(512x512) via
// V_WMMA_F32_16X16X4_F32. One wave per 16x16 output tile, 128 K-steps.
// A layout (ISA 32-bit 16x4): lanes 0-15 hold K=kOff{0,1}, lanes 16-31 K={2,3}.
// ---------------------------------------------------------------------------
__global__ __launch_bounds__(256) void k_gemm(const float* __restrict__ xw,
                                              const float* __restrict__ Wv,
                                              float* __restrict__ att)
{
    const int wave = threadIdx.x >> 5;
    const int lane = threadIdx.x & 31;
    const int tile = blockIdx.x * 8 + wave;   // 64 tiles total (grid = 8)
    const int mBase = (tile & 1) * 16;        // 2 M-tiles (rows of xw)
    const int nBase = (tile >> 1) * 16;       // 32 N-tiles (rows of W_value)
    const int mn   = lane & 15;
    const int kOff = (lane >> 4) * 2;

    v8f acc = {0.f, 0.f, 0.f, 0.f, 0.f, 0.f, 0.f, 0.f};
    const float* aRow = xw + (size_t)(mBase + mn) * DD;   // A[m, :]
    const float* bRow = Wv + (size_t)(nBase + mn) * DD;   // B[k,n] = Wv[n,k]

    for (int k = 0; k < DD; k += 4) {
        v2f a, bm;
        a.x  = aRow[k + kOff];
        a.y  = aRow[k + kOff + 1];
        bm.x = bRow[k + kOff];
        bm.y = bRow[k + kOff + 1];
        acc = __builtin_amdgcn_wmma_f32_16x16x4_f32(false, a, false, bm,
                                                    (short)0, acc, false, false);
    }

    // D layout: VGPR r -> row r (lanes 0-15) / row 8+r (lanes 16-31), col = lane&15
    const int rowOff = (lane >> 4) * 8;
    const int col    = lane & 15;
    #pragma unroll
    for (int r = 0; r < 8; ++r)
        att[(size_t)(mBase + rowOff + r) * DD + nBase + col] = acc[r];
}

// ---------------------------------------------------------------------------
// Kernel 4: char_value[b,j] = attended[b,:] . W_char[j,:]  (32x8), new_offset
// ---------------------------------------------------------------------------
__global__ __launch_bounds__(256) void k_final(const float* __restrict__ att,
                                               const float* __restrict__ Wc,
                                               const float* __restrict__ read_offset,
                                               float* __restrict__ out)
{
    const int t = threadIdx.x;       // 256 = 32*8
    const int b = t >> 3, j = t & 7;
    float s = 0.0f;
    for (int e = 0; e < DD; ++e) s += att[(size_t)b * DD + e] * Wc[j * DD + e];
    out[t] = s;                                       // char_value (first 256)
    if (t < BB) out[256 + t] = read_offset[t] + 1.0f; // new_offset (next 32)
}

extern "C" void kernel_launch(void* const* d_in, const int* in_sizes, int n_in,
                              void* d_out, int out_size, void* d_ws, size_t ws_size,
                              hipStream_t stream) {
    (void)in_sizes; (void)n_in; (void)out_size; (void)ws_size;
    const float* x           = (const float*)d_in[0];
    const int*   positions   = (const int*)d_in[1];
    const int*   anchor      = (const int*)d_in[2];
    const float* read_offset = (const float*)d_in[3];
    const int*   input_length= (const int*)d_in[4];
    const float* Wv          = (const float*)d_in[5];
    const float* Wc          = (const float*)d_in[6];

    float* out   = (float*)d_out;
    float* out_w = out + 256 + 32;     // weights region (B*S floats)

    float* ws      = (float*)d_ws;
    float* xw      = ws;                       // B*D
    float* att     = ws + BB * DD;             // B*D
    float* partial = ws + 2 * BB * DD;         // B*NCHUNK*D  (~1 MB)

    k_weights<<<BB, 256, 0, stream>>>(positions, anchor, read_offset, input_length, out_w);
    k_wsum  <<<dim3(NCHUNK, BB), 128, 0, stream>>>(x, out_w, partial);
    k_reduce<<<(BB * DD + 255) / 256, 256, 0, stream>>>(partial, xw);
    k_gemm  <<<8, 256, 0, stream>>>(xw, Wv, att);
    k_final <<<1, 256, 0, stream>>>(att, Wc, read_offset, out);
}